// ConditionalDistanceCorrelation_26010321944622
// MI455X (gfx1250) — compile-verified
//
#include <hip/hip_runtime.h>
#include <hip/hip_bf16.h>
#include <math.h>

#define N 320
#define NT 20            // 320/16 tiles
#define EPSF 1e-12f
// 1/(2*pi)^4  (d=8, sigma=1)
#define KDE_NORM (1.0f/1558.5454565440389f)

typedef float v2f __attribute__((ext_vector_type(2)));
typedef float v8f __attribute__((ext_vector_type(8)));

__device__ __forceinline__ v8f wmma_f32(v2f a, v2f b, v8f c) {
    // V_WMMA_F32_16X16X4_F32 : D = A(16x4) * B(4x16) + C(16x16), fp32
    return __builtin_amdgcn_wmma_f32_16x16x4_f32(
        false, a, false, b, (short)0, c, false, false);
}

// ---------------- Kernel 0: row squared norms of x, y, z ----------------
__global__ void norms_kernel(const float* __restrict__ x,
                             const float* __restrict__ y,
                             const float* __restrict__ z,
                             float* __restrict__ nx, float* __restrict__ ny,
                             float* __restrict__ nz) {
    int i = threadIdx.x;
    if (i >= N) return;
    float sx = 0.f, sy = 0.f, sz = 0.f;
    const float* xr = x + i * 64;
    const float* yr = y + i * 64;
    const float* zr = z + i * 8;
    for (int c = 0; c < 64; ++c) { sx += xr[c] * xr[c]; sy += yr[c] * yr[c]; }
    for (int c = 0; c < 8;  ++c) { sz += zr[c] * zr[c]; }
    nx[i] = sx; ny[i] = sy; nz[i] = sz;
}

// ------- Kernel 1: Dx, Dy distance tiles + W KDE tiles via f32 WMMA -------
__global__ __launch_bounds__(32)
void dist_kde_kernel(const float* __restrict__ x, const float* __restrict__ y,
                     const float* __restrict__ z,
                     const float* __restrict__ nx, const float* __restrict__ ny,
                     const float* __restrict__ nz,
                     float* __restrict__ Dx, float* __restrict__ Dy,
                     float* __restrict__ W) {
    const int lane = threadIdx.x;
    const int half = lane >> 4;      // 0: lanes 0-15, 1: lanes 16-31
    const int lm   = lane & 15;
    const int r0 = blockIdx.y * 16;
    const int c0 = blockIdx.x * 16;

    v8f gx = {}, gy = {}, gz = {};
    // A row for this lane (M = lm), B column (N = lm) comes from row c0+lm of x
    const float* xr = x + (r0 + lm) * 64;
    const float* xc = x + (c0 + lm) * 64;
    const float* yr = y + (r0 + lm) * 64;
    const float* yc = y + (c0 + lm) * 64;
    const float* zr = z + (r0 + lm) * 8;
    const float* zc = z + (c0 + lm) * 8;

    #pragma unroll
    for (int k = 0; k < 64; k += 4) {
        int kb = k + half * 2;
        v2f a = { xr[kb], xr[kb + 1] };
        v2f b = { xc[kb], xc[kb + 1] };
        gx = wmma_f32(a, b, gx);
    }
    #pragma unroll
    for (int k = 0; k < 64; k += 4) {
        int kb = k + half * 2;
        v2f a = { yr[kb], yr[kb + 1] };
        v2f b = { yc[kb], yc[kb + 1] };
        gy = wmma_f32(a, b, gy);
    }
    #pragma unroll
    for (int k = 0; k < 8; k += 4) {
        int kb = k + half * 2;
        v2f a = { zr[kb], zr[kb + 1] };
        v2f b = { zc[kb], zc[kb + 1] };
        gz = wmma_f32(a, b, gz);
    }

    #pragma unroll
    for (int e = 0; e < 8; ++e) {
        int i = r0 + half * 8 + e;   // C/D layout: M = e + 8*(lane/16)
        int j = c0 + lm;             // N = lane%16
        float d2x = fmaxf(nx[i] + nx[j] - 2.0f * gx[e], 0.0f);
        float d2y = fmaxf(ny[i] + ny[j] - 2.0f * gy[e], 0.0f);
        float d2z = fmaxf(nz[i] + nz[j] - 2.0f * gz[e], 0.0f);
        Dx[i * N + j] = (d2x > 0.0f) ? sqrtf(d2x) : 0.0f;
        Dy[i * N + j] = (d2y > 0.0f) ? sqrtf(d2y) : 0.0f;
        W[i * N + j]  = __expf(-0.5f * d2z) * KDE_NORM;
    }
}

// ---------------- Kernel 2: ws[k] = row sums of W ----------------
__global__ void ws_kernel(const float* __restrict__ W, float* __restrict__ ws) {
    __shared__ float s[64];
    int k = blockIdx.x;
    int t = threadIdx.x;
    float a = 0.f;
    for (int i = t; i < N; i += 64) a += W[k * N + i];
    s[t] = a;
    __syncthreads();
    for (int off = 32; off > 0; off >>= 1) {
        if (t < off) s[t] += s[t + off];
        __syncthreads();
    }
    if (t == 0) ws[k] = s[0];
}

// ---- Kernel 3: fused 5-way GEMM against W^T with shared loads ----
// Mx = Dx*W^T, My = Dy*W^T, Exy=(Dx.Dy)*W^T, Exx=(Dx.Dx)*W^T, Eyy=(Dy.Dy)*W^T
__global__ __launch_bounds__(32)
void gemm5_kernel(const float* __restrict__ Dx, const float* __restrict__ Dy,
                  const float* __restrict__ W,
                  float* __restrict__ Mx, float* __restrict__ My,
                  float* __restrict__ Exy, float* __restrict__ Exx,
                  float* __restrict__ Eyy) {
    const int lane = threadIdx.x;
    const int half = lane >> 4;
    const int lm   = lane & 15;
    const int r0 = blockIdx.y * 16;  // output rows (i)
    const int c0 = blockIdx.x * 16;  // output cols (condition index k)

    v8f cmx = {}, cmy = {}, cxy = {}, cxx = {}, cyy = {};
    const float* arx = Dx + (r0 + lm) * N;   // A rows
    const float* ary = Dy + (r0 + lm) * N;
    const float* brw = W  + (c0 + lm) * N;   // B^T rows: B[j][n] = W[n][j]

    #pragma unroll 4
    for (int kk = 0; kk < N; kk += 4) {
        int kb = kk + half * 2;
        v2f ax = { arx[kb], arx[kb + 1] };
        v2f ay = { ary[kb], ary[kb + 1] };
        v2f b  = { brw[kb], brw[kb + 1] };
        v2f axy = ax * ay;
        v2f axx = ax * ax;
        v2f ayy = ay * ay;
        cmx = wmma_f32(ax,  b, cmx);
        cmy = wmma_f32(ay,  b, cmy);
        cxy = wmma_f32(axy, b, cxy);
        cxx = wmma_f32(axx, b, cxx);
        cyy = wmma_f32(ayy, b, cyy);
    }

    #pragma unroll
    for (int e = 0; e < 8; ++e) {
        int i = r0 + half * 8 + e;
        int n = c0 + lm;
        int idx = i * N + n;
        Mx[idx]  = cmx[e];
        My[idx]  = cmy[e];
        Exy[idx] = cxy[e];
        Exx[idx] = cxx[e];
        Eyy[idx] = cyy[e];
    }
}

// ---- Kernel 4: per-k closed-form combine + mean over k ----
__global__ void finalize_kernel(const float* __restrict__ W,
                                const float* __restrict__ ws,
                                const float* __restrict__ Mx,
                                const float* __restrict__ My,
                                const float* __restrict__ Exy,
                                const float* __restrict__ Exx,
                                const float* __restrict__ Eyy,
                                float* __restrict__ out) {
    __shared__ float red[N];
    int k = threadIdx.x;

    float dcor = 0.f;
    if (k < N) {
        float t  = ws[k];
        float th = t + EPSF;
        float r  = t / th;
        float cxy = 0.f, cxx = 0.f, cyy = 0.f;
        float dX = 0.f, dY = 0.f;
        float bxy = 0.f, bxx = 0.f, byy = 0.f;
        const float* wrow = W + k * N;
        for (int i = 0; i < N; ++i) {
            float u = wrow[i];
            int idx = i * N + k;
            float mx = Mx[idx];
            float my = My[idx];
            cxy += u * Exy[idx];
            cxx += u * Exx[idx];
            cyy += u * Eyy[idx];
            dX  += u * mx;
            dY  += u * my;
            bxy += u * mx * my;
            bxx += u * mx * mx;
            byy += u * my * my;
        }
        float g1 = (2.0f * r - 4.0f) / th;
        float g2 = (2.0f - r) * (2.0f - r) / (th * th);
        float xy = cxy + bxy * g1 + dX * dY * g2;
        float xx = cxx + bxx * g1 + dX * dX * g2;
        float yy = cyy + byy * g1 + dY * dY * g2;
        dcor = xy / sqrtf(fmaxf(xx * yy, EPSF));
    }
    red[k] = dcor;
    __syncthreads();
    if (k < 160) red[k] += red[k + 160];
    __syncthreads();
    if (k < 64) {
        float a = red[k];
        for (int j = k + 64; j < 160; j += 64) a += red[j];
        red[k] = a;
    }
    __syncthreads();
    for (int off = 32; off > 0; off >>= 1) {
        if (k < off) red[k] += red[k + off];
        __syncthreads();
    }
    if (k == 0) out[0] = red[0] * (1.0f / (float)N);
}

extern "C" void kernel_launch(void* const* d_in, const int* in_sizes, int n_in,
                              void* d_out, int out_size, void* d_ws, size_t ws_size,
                              hipStream_t stream) {
    const float* x = (const float*)d_in[0];   // [320,64]
    const float* y = (const float*)d_in[1];   // [320,64]
    const float* z = (const float*)d_in[2];   // [320,8]
    float* out = (float*)d_out;

    const size_t M = (size_t)N * N;           // 102400 floats per matrix
    float* ws  = (float*)d_ws;
    float* Dx  = ws;
    float* Dy  = Dx + M;
    float* Wm  = Dy + M;
    float* Mx  = Wm + M;
    float* My  = Mx + M;
    float* Exy = My + M;
    float* Exx = Exy + M;
    float* Eyy = Exx + M;
    float* nx  = Eyy + M;
    float* ny  = nx + N;
    float* nz  = ny + N;
    float* wss = nz + N;

    norms_kernel<<<1, N, 0, stream>>>(x, y, z, nx, ny, nz);

    dim3 tiles(NT, NT);
    dist_kde_kernel<<<tiles, 32, 0, stream>>>(x, y, z, nx, ny, nz, Dx, Dy, Wm);

    ws_kernel<<<N, 64, 0, stream>>>(Wm, wss);

    gemm5_kernel<<<tiles, 32, 0, stream>>>(Dx, Dy, Wm, Mx, My, Exy, Exx, Eyy);

    finalize_kernel<<<1, N, 0, stream>>>(Wm, wss, Mx, My, Exy, Exx, Eyy, out);
}